// BuiltParametrizedOperator_89421219103203
// MI455X (gfx1250) — compile-verified
//
#include <hip/hip_runtime.h>

// RX(theta) on qubit 5 of a 12-qubit state, batched 4096 x 4096.
// out_re[j] = c*sr[j] + s*si[j^64];  out_im[j] = c*si[j] - s*sr[j^64]
// with c = cos(theta/2), s = sin(theta/2).  Pure streaming, HBM-bound.

typedef __attribute__((ext_vector_type(4))) float v4f;

static constexpr unsigned kElems      = 4096u * 4096u;   // per real/imag plane
static constexpr unsigned kImOffF4    = kElems / 4u;     // float4 offset of imag output plane
static constexpr unsigned kPairs      = kElems / 8u;     // one thread per (j, j^64) float4 pair
static constexpr unsigned kBlock      = 256u;            // 8 wave32s
static constexpr unsigned kGrid       = kPairs / kBlock; // exact cover, no tail

__device__ __forceinline__ v4f axpby(float a, v4f x, float b, v4f y) {
    // elementwise a*x + b*y ; hipcc contracts to v_fma_f32
    return a * x + b * y;
}

__global__ __launch_bounds__(kBlock) void rx_qubit5_kernel(
    const v4f* __restrict__ sr,
    const v4f* __restrict__ si,
    const float* __restrict__ theta_p,
    v4f* __restrict__ out)       // [out_re (16M floats) | out_im (16M floats)]
{
    const unsigned t = blockIdx.x * kBlock + threadIdx.x;

    // Flat float4 index: rows are 1024 float4 (aligned), column bit6 == f4 bit4.
    // Pair = (f, f+16) with bit4(f) == 0.
    const unsigned f = ((t >> 4) << 5) | (t & 15u);
    const unsigned p = f + 16u;

    const float h = 0.5f * theta_p[0];   // uniform scalar load
    const float c = __builtin_cosf(h);
    const float s = __builtin_sinf(h);

    // 4x global_load_b128 per thread; wave covers contiguous 1KB per array.
    const v4f srj = sr[f];
    const v4f srp = sr[p];
    const v4f sij = si[f];
    const v4f sip = si[p];

    const v4f rej = axpby(c, srj,  s, sip);   // out_re[j]
    const v4f rep = axpby(c, srp,  s, sij);   // out_re[j^64]
    const v4f imj = axpby(c, sij, -s, srp);   // out_im[j]
    const v4f imp = axpby(c, sip, -s, srj);   // out_im[j^64]

    // Non-temporal stores (CDNA5 TH=NT): write-once outputs must not evict the
    // 128MB of inputs that fit (and stay hot) in the 192MB L2 across replays.
    __builtin_nontemporal_store(rej, &out[f]);
    __builtin_nontemporal_store(rep, &out[p]);
    __builtin_nontemporal_store(imj, &out[kImOffF4 + f]);
    __builtin_nontemporal_store(imp, &out[kImOffF4 + p]);
}

extern "C" void kernel_launch(void* const* d_in, const int* in_sizes, int n_in,
                              void* d_out, int out_size, void* d_ws, size_t ws_size,
                              hipStream_t stream) {
    const v4f*   sr    = (const v4f*)d_in[0];     // state_re, 4096*4096 f32
    const v4f*   si    = (const v4f*)d_in[1];     // state_im, 4096*4096 f32
    const float* theta = (const float*)d_in[2];   // scalar
    v4f*         out   = (v4f*)d_out;             // out_re | out_im concatenated

    rx_qubit5_kernel<<<kGrid, kBlock, 0, stream>>>(sr, si, theta, out);
}